// MSAColumnGlobalAttention_30975304139404
// MI455X (gfx1250) — compile-verified
//
#include <hip/hip_runtime.h>
#include <math.h>

// MSAColumnGlobalAttention fused kernel for gfx1250 (MI455X).
// One workgroup (256 threads = 8 wave32) per residue column i (64 columns).
// WMMA f32 16x16x4 used twice:
//   stage A: [512x8 mnorm] x [8x16 W]  -> cols 0..7 gate-preact, 8 = k, 9 = v
//   stage B: [512x8 T]     x [8x16 wo] -> cols 0..7 output channels
// All intermediates staged in LDS; softmax masking is fully arithmetic
// (no divergent branches), only one n<8 guard around the final stores.

typedef __attribute__((ext_vector_type(2))) float v2f;
typedef __attribute__((ext_vector_type(8))) float v8f;

#define SEQ   512
#define RDIM  64
#define CDIM  8
#define HDIM  8
#define PAD   9    // mnorm row pad
#define PADP  18   // proj row pad (8*18 % 64 == 16 -> halves hit disjoint banks)

__global__ __launch_bounds__(256)
void msa_col_global_attn_kernel(const float* __restrict__ m,
                                const float* __restrict__ msa_mask,
                                const float* __restrict__ ln_w,
                                const float* __restrict__ ln_b,
                                const float* __restrict__ wq,
                                const float* __restrict__ wk,
                                const float* __restrict__ wv,
                                const float* __restrict__ wg,
                                const float* __restrict__ bg,
                                const float* __restrict__ wo,
                                const float* __restrict__ bo,
                                float* __restrict__ out)
{
    __shared__ float mnorm[SEQ * PAD];    // layernorm rows; reused for T in stage B
    __shared__ float proj[SEQ * PADP];    // WMMA D tiles (gate/k/v)
    __shared__ float msk[SEQ];
    __shared__ float accum[CDIM + 1];     // masked channel sums + mask count
    __shared__ float qh[HDIM];            // per-head global query
    __shared__ float oh[HDIM];            // per-head attention output

    const int i    = blockIdx.x;          // residue column
    const int tid  = threadIdx.x;         // 0..255
    const int lane = tid & 31;
    const int wave = tid >> 5;

    if (tid < CDIM + 1) accum[tid] = 0.f;
    __syncthreads();

    // ---------------- Phase 1: LayerNorm + masked accumulation ----------------
    float lw[CDIM], lb[CDIM];
#pragma unroll
    for (int c = 0; c < CDIM; ++c) { lw[c] = ln_w[c]; lb[c] = ln_b[c]; }

    float part[CDIM + 1];
#pragma unroll
    for (int c = 0; c < CDIM + 1; ++c) part[c] = 0.f;

    // prefetch this thread's second row while the first is processed
    __builtin_prefetch(m + ((size_t)(tid + 256) * RDIM + i) * CDIM, 0, 0);

#pragma unroll
    for (int it = 0; it < 2; ++it) {
        const int s = tid + it * 256;
        const float4* row4 = reinterpret_cast<const float4*>(
            m + ((size_t)s * RDIM + i) * CDIM);      // 32B contiguous
        const float4 xa = row4[0];
        const float4 xb = row4[1];
        float x[CDIM] = {xa.x, xa.y, xa.z, xa.w, xb.x, xb.y, xb.z, xb.w};

        float mu = 0.f;
#pragma unroll
        for (int c = 0; c < CDIM; ++c) mu += x[c];
        mu *= (1.f / CDIM);
        float var = 0.f;
#pragma unroll
        for (int c = 0; c < CDIM; ++c) { const float d = x[c] - mu; var += d * d; }
        var *= (1.f / CDIM);
        const float inv = rsqrtf(var + 1e-5f);
        const float mk  = msa_mask[(size_t)s * RDIM + i];   // 0.0 or 1.0
        msk[s] = mk;
#pragma unroll
        for (int c = 0; c < CDIM; ++c) {
            const float nv = (x[c] - mu) * inv * lw[c] + lb[c];
            mnorm[s * PAD + c] = nv;
            part[c] += mk * nv;                  // branch-free masked accumulate
        }
        part[CDIM] += mk;
    }
    // wave32 tree reduction, one LDS atomic set per wave
#pragma unroll
    for (int c = 0; c < CDIM + 1; ++c) {
#pragma unroll
        for (int off = 16; off > 0; off >>= 1)
            part[c] += __shfl_xor(part[c], off, 32);
    }
    if (lane == 0) {
#pragma unroll
        for (int c = 0; c < CDIM + 1; ++c) atomicAdd(&accum[c], part[c]);
    }
    __syncthreads();

    // ---------------- Phase 2: global query q[h] (threads 0..7) ---------------
    if (tid < HDIM) {
        const float denom = accum[CDIM] + 1e-5f;
        float acc = 0.f;
#pragma unroll
        for (int c = 0; c < CDIM; ++c)
            acc += (accum[c] / denom) * wq[tid * CDIM + c];
        qh[tid] = acc;
    }

    // ---------------- Phase 3: WMMA stage A (gate/k/v projection) -------------
    const int n    = lane & 15;   // N column of this lane
    const int half = lane >> 4;   // 0: low half, 1: high half

    // A layout (ISA 7.12.2, 32-bit A 16x4): lane L -> M = L%16, K = 2*(L/16)+j
    // B layout (symmetric):                 lane L -> K = 2*(L/16)+j, N = L%16
    v2f b_lo, b_hi;
#pragma unroll
    for (int j = 0; j < 2; ++j) {
        const int r  = 2 * half + j;       // K row within block of 4
        const int c0 = r, c4 = 4 + r;
        float w0, w4;
        if (n < 8)       { w0 = wg[n * CDIM + c0]; w4 = wg[n * CDIM + c4]; }
        else if (n == 8) { w0 = wk[c0];            w4 = wk[c4]; }
        else if (n == 9) { w0 = wv[c0];            w4 = wv[c4]; }
        else             { w0 = 0.f;               w4 = 0.f; }
        b_lo[j] = w0;
        b_hi[j] = w4;
    }

#pragma unroll
    for (int t = 0; t < 4; ++t) {
        const int s0 = (wave + 8 * t) * 16;
        v2f a_lo, a_hi;
#pragma unroll
        for (int j = 0; j < 2; ++j) {
            const int kidx = 2 * half + j;
            a_lo[j] = mnorm[(s0 + n) * PAD + kidx];
            a_hi[j] = mnorm[(s0 + n) * PAD + 4 + kidx];
        }
        v8f acc = {};
        acc = __builtin_amdgcn_wmma_f32_16x16x4_f32(false, a_lo, false, b_lo,
                                                    (short)0, acc, false, false);
        acc = __builtin_amdgcn_wmma_f32_16x16x4_f32(false, a_hi, false, b_hi,
                                                    (short)0, acc, false, false);
        // D layout: VGPR r, lane L -> M = r + 8*(L/16), N = L%16.
        // Branch-free full-tile scatter (cols 10..15 are exact zeros).
#pragma unroll
        for (int r = 0; r < 8; ++r)
            proj[(s0 + r + 8 * half) * PADP + n] = acc[r];
    }
    __syncthreads();

    // ---------------- Phase 4: per-head masked softmax (wave per head) --------
    // Fully arithmetic masking (msk is exactly 0.0/1.0): loads are
    // unconditional so no divergent exec-mask blocks are generated.
    {
        const int h = wave;
        const float q = qh[h];
        float mx = -3.0e38f;
#pragma unroll
        for (int it = 0; it < SEQ / 32; ++it) {
            const int s = lane + it * 32;
            const float kv   = proj[s * PADP + 8];
            const float bias = (1.f - msk[s]) * -3.0e38f;   // 0 if unmasked
            mx = fmaxf(mx, fmaf(q, kv, bias));
        }
#pragma unroll
        for (int off = 16; off > 0; off >>= 1)
            mx = fmaxf(mx, __shfl_xor(mx, off, 32));

        float se = 0.f, sv = 0.f;
#pragma unroll
        for (int it = 0; it < SEQ / 32; ++it) {
            const int s = lane + it * 32;
            const float kv  = proj[s * PADP + 8];
            const float vvv = proj[s * PADP + 9];
            // clamp keeps masked lanes finite (0 * exp(<=0)), no-op otherwise
            const float e = msk[s] * __expf(fminf(q * kv - mx, 0.f));
            se += e;
            sv += e * vvv;
        }
#pragma unroll
        for (int off = 16; off > 0; off >>= 1) {
            se += __shfl_xor(se, off, 32);
            sv += __shfl_xor(sv, off, 32);
        }
        if (lane == 0) oh[h] = (se > 0.f) ? (sv / se) : 0.f;
    }
    __syncthreads();

    // ---------------- Phase 5a: T[s,h] = sigmoid(gate)*o_h into mnorm ---------
    float o_r[HDIM], bg_r[HDIM];
#pragma unroll
    for (int h = 0; h < HDIM; ++h) { o_r[h] = oh[h]; bg_r[h] = bg[h]; }

#pragma unroll
    for (int it = 0; it < 2; ++it) {
        const int s = tid + it * 256;
#pragma unroll
        for (int h = 0; h < HDIM; ++h) {
            const float gp = proj[s * PADP + h] + bg_r[h];
            const float g  = 1.f / (1.f + __expf(-gp));
            mnorm[s * PAD + h] = g * o_r[h];
        }
    }
    __syncthreads();

    // ---------------- Phase 5b: WMMA stage B (output projection) --------------
    // out[s, c] = bo[c] + msk[s] * sum_h T[s,h] * wo[c,h]
    // B[k=h][n=c] = wo[c*HDIM + h] for n<8, else 0.
    v2f wb_lo, wb_hi;
#pragma unroll
    for (int j = 0; j < 2; ++j) {
        const int r = 2 * half + j;
        const float w0 = (n < 8) ? wo[n * HDIM + r]     : 0.f;
        const float w4 = (n < 8) ? wo[n * HDIM + 4 + r] : 0.f;
        wb_lo[j] = w0;
        wb_hi[j] = w4;
    }
    const float bon = (n < 8) ? bo[n] : 0.f;

#pragma unroll
    for (int t = 0; t < 4; ++t) {
        const int s0 = (wave + 8 * t) * 16;
        v2f a_lo, a_hi;
#pragma unroll
        for (int j = 0; j < 2; ++j) {
            const int kidx = 2 * half + j;
            a_lo[j] = mnorm[(s0 + n) * PAD + kidx];
            a_hi[j] = mnorm[(s0 + n) * PAD + 4 + kidx];
        }
        v8f acc = {};
        acc = __builtin_amdgcn_wmma_f32_16x16x4_f32(false, a_lo, false, wb_lo,
                                                    (short)0, acc, false, false);
        acc = __builtin_amdgcn_wmma_f32_16x16x4_f32(false, a_hi, false, wb_hi,
                                                    (short)0, acc, false, false);
        if (n < 8) {
#pragma unroll
            for (int r = 0; r < 8; ++r) {
                const int s = s0 + r + 8 * half;
                out[((size_t)s * RDIM + i) * CDIM + n] = bon + msk[s] * acc[r];
            }
        }
    }
}

extern "C" void kernel_launch(void* const* d_in, const int* in_sizes, int n_in,
                              void* d_out, int out_size, void* d_ws, size_t ws_size,
                              hipStream_t stream) {
    (void)in_sizes; (void)n_in; (void)out_size; (void)d_ws; (void)ws_size;
    const float* m        = (const float*)d_in[0];
    const float* msa_mask = (const float*)d_in[1];
    const float* ln_w     = (const float*)d_in[2];
    const float* ln_b     = (const float*)d_in[3];
    const float* wq       = (const float*)d_in[4];
    const float* wk       = (const float*)d_in[5];
    const float* wv       = (const float*)d_in[6];
    const float* wg       = (const float*)d_in[7];
    const float* bg       = (const float*)d_in[8];
    const float* wo       = (const float*)d_in[9];
    const float* bo       = (const float*)d_in[10];
    float* out            = (float*)d_out;

    dim3 grid(RDIM);     // 64 residue columns
    dim3 block(256);     // 8 wave32
    msa_col_global_attn_kernel<<<grid, block, 0, stream>>>(
        m, msa_mask, ln_w, ln_b, wq, wk, wv, wg, bg, wo, bo, out);
}